// SuperLinear_2980707304218
// MI455X (gfx1250) — compile-verified
//
#include <hip/hip_runtime.h>
#include <stdint.h>

// SuperLinear: out[b,n,o] = sum_m x[b,n,m]*w1[m,o,n] + b1[n,o]
// B=64, N=2048, M=64, O=128 -> 2048 independent 64x64x128 fp32 GEMMs.
// Memory-bound (~161 MiB @ 23.3 TB/s ~= 7us floor) -> fp32 WMMA 16x16x4,
// async global->LDS staging for X, K-pair-interleaved W for b64 B loads,
// batched B-fragment loads so LDS latency overlaps the WMMA stream.

#define B_ 64
#define N_ 2048
#define M_ 64
#define O_ 128

#define NB 4          // neurons per block (float4-coalesced w1 loads over n)
#define THREADS 512   // 16 wave32
#define GROW 288      // LDS stride (floats) per K-pair group of W (m/2 rows)
#define XROW 260      // LDS row stride (floats) for X rows (16B aligned rows)
#define SW_PER_N (32 * GROW)            // 9216 floats / neuron (32 K-pair groups)
#define SW_FLOATS (NB * SW_PER_N)       // 36864 floats
#define SX_FLOATS (B_ * XROW)           // 16640 floats
#define LDS_BYTES ((SW_FLOATS + SX_FLOATS) * 4)  // ~209 KB < 320 KB/WGP

typedef float v2f __attribute__((ext_vector_type(2)));
typedef float v8f __attribute__((ext_vector_type(8)));

__global__ __launch_bounds__(THREADS) void superlinear_wmma(
    const float* __restrict__ x, const float* __restrict__ w1,
    const float* __restrict__ b1, float* __restrict__ out)
{
  extern __shared__ float smem[];
  float* sW = smem;              // [NB][32 groups][o:128][kpair:2], stride GROW
  float* sX = smem + SW_FLOATS;  // [B_][XROW floats], first 256 used

  const int tid = threadIdx.x;
  const int n0  = blockIdx.x * NB;

  // ---- stage X asynchronously: per b, the NB neurons' rows are 256
  // contiguous floats; each lane copies 16B global -> 16B LDS (ASYNCcnt).
  for (int i = tid; i < B_ * 64; i += THREADS) {
    const int b = i >> 6;
    const int t = i & 63;  // float4 index within the 256-float chunk
    const float* src = x + (size_t)b * N_ * M_ + (size_t)n0 * M_ + t * 4;
    float* dst = sX + b * XROW + t * 4;                 // 16B aligned
    const unsigned lds_off = (unsigned)((char*)dst - (char*)smem);
    const unsigned long long ga = (unsigned long long)(uintptr_t)src;
    asm volatile("global_load_async_to_lds_b128 %0, %1, off"
                 :: "v"(lds_off), "v"(ga) : "memory");
  }

  // ---- stage W (synchronous; 4-neuron scatter): w1 is [P=M*O, N] row-major.
  // One float4 load grabs the same (m,o) for 4 consecutive neurons.
  for (int p = tid; p < M_ * O_; p += THREADS) {
    const float4 v = *(const float4*)(w1 + (size_t)p * N_ + n0);
    const int g      = p >> 8;            // m/2: K-pair group
    const int o      = p & (O_ - 1);      // p % O_
    const int parity = (p >> 7) & 1;      // m & 1
    const int base = g * GROW + o * 2 + parity;
    sW[0 * SW_PER_N + base] = v.x;
    sW[1 * SW_PER_N + base] = v.y;
    sW[2 * SW_PER_N + base] = v.z;
    sW[3 * SW_PER_N + base] = v.w;
  }

  asm volatile("s_wait_asynccnt 0x0" ::: "memory");  // drain this wave's asyncs
  __syncthreads();

  // ---- compute: wave -> (neuron j, b-tile bi); 8 o-tiles per wave.
  const int wave = tid >> 5;
  const int lane = tid & 31;
  const int j    = wave >> 2;    // neuron within block
  const int bi   = wave & 3;     // b-tile (16 rows)
  const int half = lane >> 4;    // 0/1 -> K pair {0,1} vs {2,3} (A/B layout)
  const int col  = lane & 15;
  const int n    = n0 + j;

  const float* Wj = sW + j * SW_PER_N;
  // A fragment: lane holds row (bi*16+col), K = half*2 + {0,1}
  const float* xA = sX + (bi * 16 + col) * XROW + j * M_ + half * 2;

  v8f acc[8];
#pragma unroll
  for (int oi = 0; oi < 8; ++oi) acc[oi] = (v8f)(0.0f);

  for (int k = 0; k < M_; k += 4) {
    const v2f a = *(const v2f*)(xA + k);  // ds_load_b64, conflict-free
    // B fragments: group (k/2 + half), element pair (K, K+1) adjacent in LDS.
    const float* wk = Wj + ((k >> 1) + half) * GROW + col * 2;
    v2f bf[8];
#pragma unroll
    for (int oi = 0; oi < 8; ++oi)        // issue all 8 loads up front so the
      bf[oi] = *(const v2f*)(wk + oi * 32);  // dscnt waits overlap the WMMAs
#pragma unroll
    for (int oi = 0; oi < 8; ++oi) {
      acc[oi] = __builtin_amdgcn_wmma_f32_16x16x4_f32(
          /*neg_a=*/false, a, /*neg_b=*/false, bf[oi],
          /*c_mod=*/(short)0, acc[oi], /*reuse_a=*/false, /*reuse_b=*/false);
    }
  }

  // ---- epilogue: D layout -> VGPR v holds rows v (lanes 0-15) / v+8 (16-31)
#pragma unroll
  for (int oi = 0; oi < 8; ++oi) {
    const int o = oi * 16 + col;
    const float bias = b1[n * O_ + o];
#pragma unroll
    for (int v = 0; v < 8; ++v) {
      const int brow = bi * 16 + half * 8 + v;
      out[((size_t)brow * N_ + n) * O_ + o] = acc[oi][v] + bias;
    }
  }
}

extern "C" void kernel_launch(void* const* d_in, const int* in_sizes, int n_in,
                              void* d_out, int out_size, void* d_ws, size_t ws_size,
                              hipStream_t stream) {
  const float* x  = (const float*)d_in[0];
  const float* w1 = (const float*)d_in[1];
  const float* b1 = (const float*)d_in[2];
  float* out = (float*)d_out;

  (void)hipFuncSetAttribute((const void*)superlinear_wmma,
                            hipFuncAttributeMaxDynamicSharedMemorySize,
                            LDS_BYTES);
  superlinear_wmma<<<dim3(N_ / NB), dim3(THREADS), LDS_BYTES, stream>>>(
      x, w1, b1, out);
}